// MultiHeadCrossAttention_10179072492116
// MI455X (gfx1250) — compile-verified
//
#include <hip/hip_runtime.h>
#include <math.h>

// MI455X / gfx1250 — wave32. fp32 WMMA 16x16x4 + async global->LDS staging.
// d_out = [output (4*1024*64 f32)] ++ [attn (4*8*1024*1024 f32)]
// d_ws: Qs,Ks,Vs ([B,H,L,D] f32, 8MB each) + ctx ([B,L,H*D] f32, 8MB) = 32MB

#define HH   8
#define DD   64
#define QD   512
#define BB   4
#define LL   1024
#define KC   32          // K-chunk staged per async round

typedef __attribute__((ext_vector_type(2))) float v2f;
typedef __attribute__((ext_vector_type(8))) float v8f;

// ---- CDNA5 async global->LDS (ASYNCcnt) helpers ---------------------------
__device__ __forceinline__ uint32_t lds_off(const void* p) {
    // generic shared-aperture addresses carry the LDS byte offset in [31:0]
    return (uint32_t)(uintptr_t)p;
}
__device__ __forceinline__ void async_g2l_b128(uint32_t lds, const float* g) {
    asm volatile("global_load_async_to_lds_b128 %0, %1, off"
                 :: "v"(lds), "v"(g) : "memory");
}
__device__ __forceinline__ void wait_async0() {
    asm volatile("s_wait_asynccnt 0x0" ::: "memory");
}

// ---------------------------------------------------------------------------
// out[m,n] = scale * sum_k X[m,k]*W[n,k] + bias[n]
// Block = 8 waves = 128(M) x 64(N). W chunk [64 x KC] double-buffered in LDS
// via async B128 copies (shared by all 8 waves); A-frags batch-loaded to
// registers per chunk. mode 0: out[z*so + m*Ntot + n]; mode 1: [B,H,L,D].
// ---------------------------------------------------------------------------
#define WSTR 36   // LDS row stride (floats): conflict-free, 16B-aligned rows
__global__ void __launch_bounds__(256) gemm_xwt(
        const float* __restrict__ X, const float* __restrict__ W,
        const float* __restrict__ bias, float* __restrict__ out,
        int Kdim, int Ntot, int mode, float scale,
        size_t sx, size_t sw, size_t so) {
    __shared__ float shW[2][64 * WSTR];

    const int tid  = threadIdx.x;
    const int wave = tid >> 5;
    const int lane = tid & 31;
    const int l16  = lane & 15;
    const int half = lane >> 4;                // A/B frag K-halves per ISA layout
    const int m0   = (blockIdx.x * 8 + wave) * 16;
    const int n0   = blockIdx.y * 64;
    const int z    = blockIdx.z;

    const float* Xr = X + sx * z + (size_t)(m0 + l16) * Kdim + 2 * half;
    const float* Wz = W + sw * z + (size_t)n0 * Kdim;

    // staging map: thread -> row tid/4 (0..63), 8 floats at col (tid%4)*8
    const int    srow = tid >> 2;
    const int    scol = (tid & 3) * 8;
    const float* gW   = Wz + (size_t)srow * Kdim + scol;
    const uint32_t lW[2] = { lds_off(&shW[0][srow * WSTR + scol]),
                             lds_off(&shW[1][srow * WSTR + scol]) };

    const int nCh = Kdim / KC;
    async_g2l_b128(lW[0],      gW);
    async_g2l_b128(lW[0] + 16, gW + 4);
    wait_async0();
    __syncthreads();

    v8f acc[4] = {{}, {}, {}, {}};
    for (int c = 0; c < nCh; ++c) {
        if (c + 1 < nCh) {                      // prefetch next chunk (overlaps)
            const float*   g = gW + (size_t)(c + 1) * KC;
            const uint32_t l = lW[(c + 1) & 1];
            async_g2l_b128(l,      g);
            async_g2l_b128(l + 16, g + 4);
        }
        const float* sW = &shW[c & 1][0];
        const float* Xc = Xr + (size_t)c * KC;

        v2f areg[KC / 4];
#pragma unroll
        for (int s = 0; s < KC / 4; ++s)        // batch A loads (issue ahead)
            areg[s] = *(const v2f*)(Xc + 4 * s);
#pragma unroll
        for (int s = 0; s < KC / 4; ++s) {
            const int ko = 4 * s + 2 * half;
#pragma unroll
            for (int t = 0; t < 4; ++t) {
                v2f b = *(const v2f*)(sW + (t * 16 + l16) * WSTR + ko);
                acc[t] = __builtin_amdgcn_wmma_f32_16x16x4_f32(
                    false, areg[s], false, b, (short)0, acc[t], false, false);
            }
        }
        wait_async0();
        __syncthreads();
    }

    if (mode == 0) {
        float* Ob = out + so * z;
#pragma unroll
        for (int t = 0; t < 4; ++t) {
            const int   n  = n0 + t * 16 + l16;
            const float bv = bias ? bias[n] : 0.0f;
#pragma unroll
            for (int r = 0; r < 8; ++r) {
                const int m = m0 + r + 8 * half;
                Ob[(size_t)m * Ntot + n] = acc[t][r] * scale + bv;
            }
        }
    } else {  // head-major [B,H,L,D]; m = b*L+l, n = h*64+d
#pragma unroll
        for (int t = 0; t < 4; ++t) {
            const int   n  = n0 + t * 16 + l16;
            const int   h  = n >> 6, d = n & 63;
            const float bv = bias ? bias[n] : 0.0f;
#pragma unroll
            for (int r = 0; r < 8; ++r) {
                const int m = m0 + r + 8 * half;
                const int b = m >> 10, l = m & 1023;
                out[(((size_t)(b * HH + h)) * LL + l) * DD + d] = acc[t][r] * scale + bv;
            }
        }
    }
}

// ---------------------------------------------------------------------------
// In-place row softmax over 1024-wide rows of attn (one block per row).
// ---------------------------------------------------------------------------
__global__ void __launch_bounds__(256) softmax_rows(float* __restrict__ attn) {
    __shared__ float red[256];
    float* p = attn + (size_t)blockIdx.x * LL;
    const int t = threadIdx.x;

    float x0 = p[t], x1 = p[t + 256], x2 = p[t + 512], x3 = p[t + 768];
    float m = fmaxf(fmaxf(x0, x1), fmaxf(x2, x3));
    red[t] = m; __syncthreads();
    for (int s = 128; s > 0; s >>= 1) { if (t < s) red[t] = fmaxf(red[t], red[t + s]); __syncthreads(); }
    m = red[0]; __syncthreads();

    x0 = __expf(x0 - m); x1 = __expf(x1 - m); x2 = __expf(x2 - m); x3 = __expf(x3 - m);
    red[t] = x0 + x1 + x2 + x3; __syncthreads();
    for (int s = 128; s > 0; s >>= 1) { if (t < s) red[t] += red[t + s]; __syncthreads(); }
    const float inv = 1.0f / red[0];

    p[t] = x0 * inv; p[t + 256] = x1 * inv; p[t + 512] = x2 * inv; p[t + 768] = x3 * inv;
}

// ---------------------------------------------------------------------------
// ctx[b, kpos, h*64+d] = sum_q attn[b,h,q,kpos] * Vs[b,h,q,d]   (attn^T @ V)
// Block = 8 waves = 128(kpos) x 64(d). attn chunk [32q x 128k] (coalesced
// 512B row runs) and V chunk [32q x 64d] double-buffered in LDS via async
// B128; the transposed A-frag read happens in LDS, conflict-free.
// ---------------------------------------------------------------------------
#define ASTR 136
#define VSTR 72
__global__ void __launch_bounds__(256) attnT_v(const float* __restrict__ attn,
                                               const float* __restrict__ Vs,
                                               float* __restrict__ ctx) {
    __shared__ float shA[2][32 * ASTR];
    __shared__ float shV[2][32 * VSTR];

    const int tid  = threadIdx.x;
    const int wave = tid >> 5;
    const int lane = tid & 31;
    const int l16  = lane & 15;
    const int half = lane >> 4;
    const int z    = blockIdx.z;               // b*H + h
    const int mB   = blockIdx.x * 128;         // kpos block base
    const int mloc = wave * 16;

    const float* Ab = attn + (size_t)z * LL * LL + mB;   // + q*LL + 0..127
    const float* Vb = Vs   + (size_t)z * LL * DD;

    // staging maps (all 16B aligned, fully coalesced)
    const int arow = tid >> 3, acol = (tid & 7) * 16;    // 4x B128 per thread
    const int vrow = tid >> 3, vcol = (tid & 7) * 8;     // 2x B128 per thread
    const float* gA = Ab + (size_t)arow * LL + acol;
    const float* gV = Vb + (size_t)vrow * DD + vcol;
    const uint32_t lA[2] = { lds_off(&shA[0][arow * ASTR + acol]),
                             lds_off(&shA[1][arow * ASTR + acol]) };
    const uint32_t lV[2] = { lds_off(&shV[0][vrow * VSTR + vcol]),
                             lds_off(&shV[1][vrow * VSTR + vcol]) };

    auto issue = [&](int chunk, int buf) {
        const float* ga = gA + (size_t)chunk * 32 * LL;
#pragma unroll
        for (int i = 0; i < 4; ++i)
            async_g2l_b128(lA[buf] + 16 * i, ga + 4 * i);
        const float* gv = gV + (size_t)chunk * 32 * DD;
        async_g2l_b128(lV[buf],      gv);
        async_g2l_b128(lV[buf] + 16, gv + 4);
    };

    issue(0, 0);
    wait_async0();
    __syncthreads();

    v8f acc[4] = {{}, {}, {}, {}};
    const int nCh = LL / 32;
    for (int c = 0; c < nCh; ++c) {
        if (c + 1 < nCh) issue(c + 1, (c + 1) & 1);
        const float* sA = &shA[c & 1][0];
        const float* sV = &shV[c & 1][0];
#pragma unroll
        for (int q0 = 0; q0 < 32; q0 += 4) {
            const int qq = q0 + 2 * half;
            v2f a;                                    // A[m][q] = attn[q][m]
            a[0] = sA[(qq + 0) * ASTR + mloc + l16];
            a[1] = sA[(qq + 1) * ASTR + mloc + l16];
#pragma unroll
            for (int t = 0; t < 4; ++t) {
                v2f b;                                // B[q][d] = V[q][d]
                b[0] = sV[(qq + 0) * VSTR + t * 16 + l16];
                b[1] = sV[(qq + 1) * VSTR + t * 16 + l16];
                acc[t] = __builtin_amdgcn_wmma_f32_16x16x4_f32(
                    false, a, false, b, (short)0, acc[t], false, false);
            }
        }
        wait_async0();
        __syncthreads();
    }

    const int b = z >> 3, h = z & 7;
    float* Cb = ctx + (size_t)b * LL * (HH * DD) + (size_t)h * DD;
#pragma unroll
    for (int t = 0; t < 4; ++t)
#pragma unroll
        for (int r = 0; r < 8; ++r)
            Cb[(size_t)(mB + mloc + r + 8 * half) * (HH * DD) + t * 16 + l16] = acc[t][r];
}

extern "C" void kernel_launch(void* const* d_in, const int* in_sizes, int n_in,
                              void* d_out, int out_size, void* d_ws, size_t ws_size,
                              hipStream_t stream) {
    const float* q  = (const float*)d_in[0];
    const float* k  = (const float*)d_in[1];
    const float* v  = (const float*)d_in[2];
    const float* wq = (const float*)d_in[3];
    const float* bq = (const float*)d_in[4];
    const float* wk = (const float*)d_in[5];
    const float* bk = (const float*)d_in[6];
    const float* wv = (const float*)d_in[7];
    const float* bv = (const float*)d_in[8];
    const float* wp = (const float*)d_in[9];
    const float* bp = (const float*)d_in[10];

    float* out  = (float*)d_out;                       // [4,1024,64]
    float* attn = out + (size_t)BB * LL * DD;          // [4,8,1024,1024]

    const size_t headSz = (size_t)BB * HH * LL * DD;   // 2 Mi floats
    float* Qs  = (float*)d_ws;
    float* Ks  = Qs + headSz;
    float* Vs  = Ks + headSz;
    float* ctx = Vs + headSz;                          // [4,1024,512]

    // 1) Projections: [4096x512] @ W^T -> head-major [B,H,L,D]
    gemm_xwt<<<dim3(32, 8, 1), 256, 0, stream>>>(q, wq, bq, Qs, QD, QD, 1, 1.0f, 0, 0, 0);
    gemm_xwt<<<dim3(32, 8, 1), 256, 0, stream>>>(k, wk, bk, Ks, QD, QD, 1, 1.0f, 0, 0, 0);
    gemm_xwt<<<dim3(32, 8, 1), 256, 0, stream>>>(v, wv, bv, Vs, QD, QD, 1, 1.0f, 0, 0, 0);

    // 2) scores = 0.5 * Q @ K^T per (b,h), written straight into attn output
    gemm_xwt<<<dim3(8, 16, BB * HH), 256, 0, stream>>>(
        Qs, Ks, nullptr, attn, DD, LL, 0, 0.5f,
        (size_t)LL * DD, (size_t)LL * DD, (size_t)LL * LL);

    // 3) softmax over k, in place (attn is an output of the module)
    softmax_rows<<<BB * HH * LL, 256, 0, stream>>>(attn);

    // 4) context = attn^T @ V  -> ctx as [B, L, H*D]
    attnT_v<<<dim3(8, 1, BB * HH), 256, 0, stream>>>(attn, Vs, ctx);

    // 5) output = ctx @ wp^T + bp  -> [4096 x 64]
    gemm_xwt<<<dim3(32, 1, 1), 256, 0, stream>>>(ctx, wp, bp, out, QD, DD, 0, 1.0f, 0, 0, 0);
}